// SimpleModel_41566693491383
// MI455X (gfx1250) — compile-verified
//
#include <hip/hip_runtime.h>
#include <hip/hip_bf16.h>
#include <math.h>

typedef __attribute__((ext_vector_type(2))) float v2f;
typedef __attribute__((ext_vector_type(8))) float v8f;

#define NB   229   // N_BINS
#define MD   128   // MODEL
#define OUTD 88    // OUT
#define TT   2048
#define BB   4
#define BT   (BB * TT)   // 8192
#define WLEN 61
#define WSZ  30
#define KMAIN 228  // 57 guard-free chunks of 4 cover k = 0..227; tail k = 228

// ---------------------------------------------------------------------------
// Kernel A: H = spec @ W_h + b_attn ; E = spec @ W_e
// One wave per 16x16 tile. Guard-free main K loop (57 chunks), explicit tail.
// ---------------------------------------------------------------------------
__global__ void proj_wmma(const float* __restrict__ spec,
                          const float* __restrict__ W_h,
                          const float* __restrict__ W_e,
                          const float* __restrict__ b_attn,
                          float* __restrict__ H,
                          float* __restrict__ E) {
    const int lane = threadIdx.x;        // 0..31
    const int half = lane >> 4;          // A/B vgpr K-half select
    const int r    = lane & 15;
    const int row  = blockIdx.x * 16 + r;   // A-matrix row for this lane
    const int col  = blockIdx.y * 16 + r;   // B/N column for this lane

    const float bv = b_attn[col];
    v8f ch = {bv, bv, bv, bv, bv, bv, bv, bv};  // bias folded into C
    v8f ce = {0.f, 0.f, 0.f, 0.f, 0.f, 0.f, 0.f, 0.f};

    const float* __restrict__ ap = spec + (size_t)row * NB + 2 * half;
    const float* __restrict__ hp = W_h + (size_t)(2 * half) * MD + col;
    const float* __restrict__ ep = W_e + (size_t)(2 * half) * MD + col;

    // k = 0..227: every load in-bounds, zero control flow in the loop body.
    for (int kb = 0; kb < KMAIN; kb += 4) {
        v2f a, bh, be;
        a.x  = ap[0];   a.y  = ap[1];
        bh.x = hp[0];   bh.y = hp[MD];
        be.x = ep[0];   be.y = ep[MD];
        ch = __builtin_amdgcn_wmma_f32_16x16x4_f32(false, a, false, bh,
                                                   (short)0, ch, false, false);
        ce = __builtin_amdgcn_wmma_f32_16x16x4_f32(false, a, false, be,
                                                   (short)0, ce, false, false);
        ap += 4;
        hp += 4 * MD;
        ep += 4 * MD;
    }

    // Tail chunk: only k = 228 is live (half==0, vgpr0). Zero invalid A
    // entries via cndmask; B loads are unconditional (multiplied by zero A).
    {
        const float at = spec[(size_t)row * NB + KMAIN];   // always in-bounds
        v2f a, bh, be;
        a.x  = half ? 0.f : at;
        a.y  = 0.f;
        bh.x = W_h[KMAIN * MD + col];  bh.y = 0.f;
        be.x = W_e[KMAIN * MD + col];  be.y = 0.f;
        ch = __builtin_amdgcn_wmma_f32_16x16x4_f32(false, a, false, bh,
                                                   (short)0, ch, false, false);
        ce = __builtin_amdgcn_wmma_f32_16x16x4_f32(false, a, false, be,
                                                   (short)0, ce, false, false);
    }

    // C/D layout: vgpr v -> M = v + 8*half, N = lane&15
    for (int v = 0; v < 8; ++v) {
        const int gr = blockIdx.x * 16 + v + 8 * half;
        H[(size_t)gr * MD + col] = ch[v];
        E[(size_t)gr * MD + col] = ce[v];
    }
}

// ---------------------------------------------------------------------------
// Kernel B: scores = v . tanh(H[t] + E[t+w-30]), softmax over w,
//           weighted[f] = sum_w a_w * spec[t+w-30, f], emit a.
// One 64-thread block per (b,t). float4 E loads (rows are 512B aligned).
// ---------------------------------------------------------------------------
__global__ void attn_softmax(const float* __restrict__ spec,
                             const float* __restrict__ v_w,
                             const float* __restrict__ H,
                             const float* __restrict__ E,
                             float* __restrict__ weighted,
                             float* __restrict__ a_out) {
    const int bt  = blockIdx.x;          // b*T + t
    const int t   = bt % TT;
    const int tid = threadIdx.x;         // 0..63

    __shared__ float sH[MD];
    __shared__ float sV[MD];
    __shared__ float sA[WLEN];

    sH[tid]      = H[(size_t)bt * MD + tid];
    sH[tid + 64] = H[(size_t)bt * MD + tid + 64];
    sV[tid]      = v_w[tid];
    sV[tid + 64] = v_w[tid + 64];
    __syncthreads();

    if (tid < WLEN) {
        const int s = t + tid - WSZ;
        float acc = 0.f;
        if (s >= 0 && s < TT) {
            const float4* __restrict__ Er4 =
                (const float4*)(E + (size_t)(bt - t + s) * MD);
            for (int q = 0; q < MD / 4; ++q) {
                const float4 e = Er4[q];
                const int m = 4 * q;
                acc += sV[m + 0] * tanhf(sH[m + 0] + e.x);
                acc += sV[m + 1] * tanhf(sH[m + 1] + e.y);
                acc += sV[m + 2] * tanhf(sH[m + 2] + e.z);
                acc += sV[m + 3] * tanhf(sH[m + 3] + e.w);
            }
        } else {
            for (int m = 0; m < MD; ++m)
                acc += sV[m] * tanhf(sH[m]);
        }
        sA[tid] = acc;
    }
    __syncthreads();

    if (tid == 0) {
        float mx = sA[0];
        for (int w = 1; w < WLEN; ++w) mx = fmaxf(mx, sA[w]);
        float sum = 0.f;
        for (int w = 0; w < WLEN; ++w) {
            const float e = __expf(sA[w] - mx);
            sA[w] = e;
            sum += e;
        }
        const float inv = 1.f / sum;
        for (int w = 0; w < WLEN; ++w) sA[w] *= inv;
    }
    __syncthreads();

    if (tid < WLEN) a_out[(size_t)bt * WLEN + tid] = sA[tid];

    const int wlo = (t >= WSZ) ? 0 : (WSZ - t);
    const int whi = ((TT - 1 - t) >= WSZ) ? (WLEN - 1) : (WSZ + TT - 1 - t);
    const float* __restrict__ specb = spec + (size_t)(bt - t) * NB; // batch base

    for (int f = tid; f < NB; f += 64) {
        float acc = 0.f;
        const float* __restrict__ sp = specb + (size_t)(t + wlo - WSZ) * NB + f;
        for (int w = wlo; w <= whi; ++w) {
            acc += sA[w] * sp[0];
            sp += NB;
        }
        weighted[(size_t)bt * NB + f] = acc;
    }
}

// ---------------------------------------------------------------------------
// Kernel C: out = sigmoid(weighted @ W1 + b1), (8192x229)*(229x88).
// Guard-free K loop; out-of-range columns use clamped loads and are simply
// never stored (garbage accumulator lanes are dead).
// ---------------------------------------------------------------------------
__global__ void final_wmma(const float* __restrict__ Wt,
                           const float* __restrict__ W1,
                           const float* __restrict__ b1,
                           float* __restrict__ out) {
    const int lane = threadIdx.x;
    const int half = lane >> 4;
    const int r    = lane & 15;
    const int row  = blockIdx.x * 16 + r;
    const int col  = blockIdx.y * 16 + r;
    const bool cok = (col < OUTD);
    const int ccol = cok ? col : (OUTD - 1);   // clamped, always valid

    const float bv = b1[ccol];
    v8f acc = {bv, bv, bv, bv, bv, bv, bv, bv};

    const float* __restrict__ ap = Wt + (size_t)row * NB + 2 * half;
    const float* __restrict__ wp = W1 + (size_t)(2 * half) * OUTD + ccol;

    for (int kb = 0; kb < KMAIN; kb += 4) {
        v2f a, b;
        a.x = ap[0];  a.y = ap[1];
        b.x = wp[0];  b.y = wp[OUTD];
        acc = __builtin_amdgcn_wmma_f32_16x16x4_f32(false, a, false, b,
                                                    (short)0, acc, false, false);
        ap += 4;
        wp += 4 * OUTD;
    }

    // Tail: k = 228 live only for half==0 vgpr0 of A.
    {
        const float at = Wt[(size_t)row * NB + KMAIN];
        v2f a, b;
        a.x = half ? 0.f : at;
        a.y = 0.f;
        b.x = W1[KMAIN * OUTD + ccol];
        b.y = 0.f;
        acc = __builtin_amdgcn_wmma_f32_16x16x4_f32(false, a, false, b,
                                                    (short)0, acc, false, false);
    }

    if (cok) {
        for (int v = 0; v < 8; ++v) {
            const int gr = blockIdx.x * 16 + v + 8 * half;
            const float x = acc[v];
            out[(size_t)gr * OUTD + col] = 1.f / (1.f + __expf(-x));
        }
    }
}

// ---------------------------------------------------------------------------
extern "C" void kernel_launch(void* const* d_in, const int* in_sizes, int n_in,
                              void* d_out, int out_size, void* d_ws, size_t ws_size,
                              hipStream_t stream) {
    const float* spec   = (const float*)d_in[0];  // (4,2048,229)
    const float* W_h    = (const float*)d_in[1];  // (229,128)
    const float* W_e    = (const float*)d_in[2];  // (229,128)
    const float* b_attn = (const float*)d_in[3];  // (128,)
    const float* v_w    = (const float*)d_in[4];  // (128,)
    const float* W1     = (const float*)d_in[5];  // (229,88)
    const float* b1     = (const float*)d_in[6];  // (88,)

    float* out_pred = (float*)d_out;                       // (4,2048,88)
    float* out_a    = out_pred + (size_t)BT * OUTD;        // (4,2048,1,61)

    float* H  = (float*)d_ws;                              // BT*128 f32
    float* E  = H + (size_t)BT * MD;                       // BT*128 f32
    float* Wt = E + (size_t)BT * MD;                       // BT*229 f32

    proj_wmma<<<dim3(BT / 16, MD / 16), 32, 0, stream>>>(spec, W_h, W_e, b_attn, H, E);
    attn_softmax<<<BT, 64, 0, stream>>>(spec, v_w, H, E, Wt, out_a);
    final_wmma<<<dim3(BT / 16, (OUTD + 15) / 16), 32, 0, stream>>>(Wt, W1, b1, out_pred);
}